// KWinnersTakeAll_18442589570222
// MI455X (gfx1250) — compile-verified
//
#include <hip/hip_runtime.h>

#define E_DIM   8192
#define K_ACT   410       // ceil(0.05 * 8192)
#define THREADS 256
#define NBINS   256
#define CHUNKS  (E_DIM / 4 / THREADS)   // 8 x uint4 per thread

typedef int   v4i __attribute__((ext_vector_type(4)));
typedef float v4f __attribute__((ext_vector_type(4)));
typedef __attribute__((address_space(1))) v4i __gv4i;  // global int4
typedef __attribute__((address_space(3))) v4i __lv4i;  // LDS int4

#if __has_builtin(__builtin_amdgcn_global_load_async_to_lds_b128)
#define HAVE_ASYNC_B128 1
#else
#define HAVE_ASYNC_B128 0
#endif

__device__ __forceinline__ void wait_asynccnt0() {
#if __has_builtin(__builtin_amdgcn_s_wait_asynccnt)
  __builtin_amdgcn_s_wait_asynccnt(0);
#else
  asm volatile("s_wait_asynccnt 0" ::: "memory");
#endif
}

// 16 bytes global -> LDS via the gfx1250 async-to-LDS DMA path (ASYNCcnt).
__device__ __forceinline__ void async_copy16(const float* g, unsigned* l) {
#if HAVE_ASYNC_B128
  __builtin_amdgcn_global_load_async_to_lds_b128((__gv4i*)g, (__lv4i*)l, 0, 0);
#else
  unsigned lds_off = (unsigned)(unsigned long long)(__lv4i*)l;
  asm volatile("global_load_async_to_lds_b128 %0, %1, off"
               :: "v"(lds_off), "v"(g) : "memory");
#endif
}

__device__ __forceinline__ void nt_store_f4(v4f* p, v4f v) {
#if __has_builtin(__builtin_nontemporal_store)
  __builtin_nontemporal_store(v, p);
#else
  *p = v;
#endif
}

// Order-preserving float -> uint key (larger float => larger key).
__device__ __forceinline__ unsigned f2key(unsigned u) {
  return u ^ ((unsigned)((int)u >> 31) | 0x80000000u);
}
__device__ __forceinline__ float key2f(unsigned k) {
  unsigned u = k ^ ((~(unsigned)((int)k >> 31)) | 0x80000000u);
  return __uint_as_float(u);
}

__global__ __launch_bounds__(THREADS)
void kwta_radix_select_kernel(const float* __restrict__ x, float* __restrict__ out)
{
  __shared__ unsigned keys[E_DIM];     // 32 KB: raw floats, then keys in place
  __shared__ unsigned histA[NBINS];
  __shared__ unsigned histB[NBINS];
  __shared__ unsigned sPrefA, sMaskA, sRemA;
  __shared__ unsigned sPrefB, sMaskB, sRemB;
  __shared__ float    sThresh;

  const int    tid    = threadIdx.x;
  const size_t rowOff = (size_t)blockIdx.x * (size_t)E_DIM;
  const float* gx     = x + rowOff;

  // Resolve one radix level: wave 0 picks rank-A's bin, wave 1 rank-B's.
  // 8 descending bins per lane + shfl_up scan of chunk sums.
  auto resolve = [&](int shift, bool same) {
    const int wave = tid >> 5, lane = tid & 31;
    if (wave < 2) {
      const unsigned* h   = (wave == 0) ? histA : (same ? histA : histB);
      const unsigned  rem = (wave == 0) ? sRemA : sRemB;
      const unsigned  p0  = (wave == 0) ? sPrefA : sPrefB;
      const unsigned  m0  = (wave == 0) ? sMaskA : sMaskB;
      unsigned c[8]; unsigned s = 0u;
      const int base = 255 - lane * 8;                  // descending chunks
      #pragma unroll
      for (int j = 0; j < 8; ++j) { c[j] = h[base - j]; s += c[j]; }
      unsigned pre = s;                                 // inclusive scan of chunk sums
      #pragma unroll
      for (int off = 1; off < 32; off <<= 1) {
        unsigned tsum = __shfl_up(pre, off, 32);
        if (lane >= off) pre += tsum;
      }
      const unsigned excl = pre - s;                    // count in bins above my chunk
      if (excl < rem && rem <= excl + s) {              // unique crossing lane
        unsigned acc = excl;
        #pragma unroll
        for (int j = 0; j < 8; ++j) {
          if (acc + c[j] >= rem) {
            const unsigned b  = (unsigned)(base - j);
            const unsigned np = p0 | (b << shift);
            const unsigned nm = m0 | (0xFFu << shift);
            if (wave == 0) { sPrefA = np; sMaskA = nm; sRemA = rem - acc; }
            else           { sPrefB = np; sMaskB = nm; sRemB = rem - acc; }
            break;
          }
          acc += c[j];
        }
      }
    }
  };

  // ---- Phase 0: async DMA the row into LDS (2048 x b128, 8 per thread) ----
  #pragma unroll
  for (int k = 0; k < CHUNKS; ++k) {
    const int c = tid + k * THREADS;
    async_copy16(gx + 4 * c, &keys[4 * c]);
  }
  histA[tid] = 0u;
  if (tid == 0) {
    sPrefA = 0u; sMaskA = 0u; sRemA = K_ACT;
    sPrefB = 0u; sMaskB = 0u; sRemB = K_ACT + 1;
  }
  wait_asynccnt0();
  __syncthreads();

  // ---- Phase 1: in-place float -> key, fused with level-0 histogram ----
  #pragma unroll
  for (int k = 0; k < CHUNKS; ++k) {
    const int c = tid + k * THREADS;
    uint4 v = reinterpret_cast<uint4*>(keys)[c];
    v.x = f2key(v.x); v.y = f2key(v.y); v.z = f2key(v.z); v.w = f2key(v.w);
    reinterpret_cast<uint4*>(keys)[c] = v;
    atomicAdd(&histA[v.x >> 24], 1u);
    atomicAdd(&histA[v.y >> 24], 1u);
    atomicAdd(&histA[v.z >> 24], 1u);
    atomicAdd(&histA[v.w >> 24], 1u);
  }
  __syncthreads();
  resolve(24, /*same=*/true);          // both ranks resolved from histA
  __syncthreads();

  // ---- Phase 2: radix-256 levels 1..3 for descending ranks K and K+1 ----
  for (int level = 1; level < 4; ++level) {
    const int      shift = 24 - 8 * level;
    const unsigned pA = sPrefA, mA = sMaskA;
    const unsigned pB = sPrefB, mB = sMaskB;
    const bool     same = (pA == pB) && (mA == mB);  // uniform across block

    histA[tid] = 0u;
    if (!same) histB[tid] = 0u;
    __syncthreads();

    #pragma unroll
    for (int k = 0; k < CHUNKS; ++k) {
      const int   c = tid + k * THREADS;
      const uint4 v = reinterpret_cast<const uint4*>(keys)[c];
      #pragma unroll
      for (int e = 0; e < 4; ++e) {
        const unsigned key = (e == 0) ? v.x : (e == 1) ? v.y : (e == 2) ? v.z : v.w;
        if ((key & mA) == pA) atomicAdd(&histA[(key >> shift) & (NBINS - 1)], 1u);
        if (!same && ((key & mB) == pB)) atomicAdd(&histB[(key >> shift) & (NBINS - 1)], 1u);
      }
    }
    __syncthreads();
    resolve(shift, same);
    __syncthreads();
  }

  if (tid == 0) sThresh = 0.5f * (key2f(sPrefA) + key2f(sPrefB));
  __syncthreads();

  // ---- Phase 3: vectorized non-temporal mask write ----
  const float t = sThresh;
  float* orow = out + rowOff;
  #pragma unroll
  for (int k = 0; k < CHUNKS; ++k) {
    const int   c  = tid + k * THREADS;
    const uint4 kv = reinterpret_cast<const uint4*>(keys)[c];
    v4f o;
    o.x = key2f(kv.x) > t ? 1.0f : 0.0f;
    o.y = key2f(kv.y) > t ? 1.0f : 0.0f;
    o.z = key2f(kv.z) > t ? 1.0f : 0.0f;
    o.w = key2f(kv.w) > t ? 1.0f : 0.0f;
    nt_store_f4(reinterpret_cast<v4f*>(orow) + c, o);
  }
}

extern "C" void kernel_launch(void* const* d_in, const int* in_sizes, int n_in,
                              void* d_out, int out_size, void* d_ws, size_t ws_size,
                              hipStream_t stream) {
  (void)n_in; (void)d_ws; (void)ws_size; (void)out_size;
  const float* x   = (const float*)d_in[0];
  float*       out = (float*)d_out;
  const int    rows = in_sizes[0] / E_DIM;   // 8192
  hipLaunchKernelGGL(kwta_radix_select_kernel, dim3(rows), dim3(THREADS), 0, stream,
                     x, out);
}